// SAGEConv_36885179138430
// MI455X (gfx1250) — compile-verified
//
#include <hip/hip_runtime.h>
#include <math.h>

typedef float v2f __attribute__((ext_vector_type(2)));
typedef float v8f __attribute__((ext_vector_type(8)));

#define D 256
#define LDS_PITCH 260   // 256 + 4 pad: conflict-free 16-row x float2 fragment reads

// ---------------- workspace zeroing ----------------
__global__ void sage_zero_kernel(float* __restrict__ p, int n) {
    int i = blockIdx.x * blockDim.x + threadIdx.x;
    int stride = gridDim.x * blockDim.x;
    for (; i < n; i += stride) p[i] = 0.0f;
}

// ---------------- in-degree counts ----------------
__global__ void sage_count_kernel(const long long* __restrict__ ei,
                                  float* __restrict__ counts, int E) {
    int i = blockIdx.x * blockDim.x + threadIdx.x;
    int stride = gridDim.x * blockDim.x;
    for (; i < E; i += stride) {
        int dst = (int)ei[E + i];
        unsafeAtomicAdd(&counts[dst], 1.0f);
    }
}

// ---------------- edge gather + scatter-add ----------------
// One wave per edge: edge indices are wave-uniform -> scalar loads.
// Each lane moves 8 contiguous floats (2x float4 loads, 8 f32 atomics).
__global__ __launch_bounds__(256)
void sage_scatter_kernel(const long long* __restrict__ ei,
                         const float* __restrict__ x,
                         float* __restrict__ sums, int E) {
    const int lane = threadIdx.x & 31;
    const int wid  = threadIdx.x >> 5;
    const int wavesPerBlock = blockDim.x >> 5;
    long long e = (long long)blockIdx.x * wavesPerBlock + wid;
    const long long estride = (long long)gridDim.x * wavesPerBlock;
    for (; e < E; e += estride) {
        const int src = (int)ei[e];
        const int dst = (int)ei[E + e];
        const float4* xs = (const float4*)(x + (size_t)src * D);
        float* sd = sums + (size_t)dst * D;
        float4 a = xs[lane * 2 + 0];
        float4 b = xs[lane * 2 + 1];
        const int c = lane * 8;
        unsafeAtomicAdd(&sd[c + 0], a.x);
        unsafeAtomicAdd(&sd[c + 1], a.y);
        unsafeAtomicAdd(&sd[c + 2], a.z);
        unsafeAtomicAdd(&sd[c + 3], a.w);
        unsafeAtomicAdd(&sd[c + 4], b.x);
        unsafeAtomicAdd(&sd[c + 5], b.y);
        unsafeAtomicAdd(&sd[c + 6], b.z);
        unsafeAtomicAdd(&sd[c + 7], b.w);
    }
}

// ---------------- fused dual GEMM + bias (WMMA f32 16x16x4) ----------------
// Block = 128 threads = 4 waves, owns 16 node rows x all 256 output columns.
// Wave w owns columns [w*64, w*64+64): 4 N-tiles, 2 GEMMs => 8 v8f accumulators.
__global__ __launch_bounds__(128)
void sage_gemm_kernel(const float* __restrict__ x,
                      const float* __restrict__ sums,
                      const float* __restrict__ counts,
                      const float* __restrict__ Wl,   // [256,256] row-major [out,in]
                      const float* __restrict__ Wr,
                      const float* __restrict__ bias,
                      float* __restrict__ out) {
    __shared__ float ldsX[16][LDS_PITCH];  // x rows
    __shared__ float ldsA[16][LDS_PITCH];  // aggr rows = sums * inv_count
    __shared__ float invc[16];

    const int tid = threadIdx.x;
    const int node0 = blockIdx.x * 16;

    if (tid < 16) {
        float c = counts[node0 + tid];
        invc[tid] = 1.0f / fmaxf(c, 1.0f);
    }
    __syncthreads();

    // cooperative stage: 16 rows x 256 cols = 1024 float4 per array (coalesced)
    for (int i = tid; i < 1024; i += 128) {
        const int r = i >> 6;          // 64 float4 per row
        const int c = (i & 63) << 2;
        const float4 xv = *(const float4*)(x + (size_t)(node0 + r) * D + c);
        *(float4*)&ldsX[r][c] = xv;
        float4 sv = *(const float4*)(sums + (size_t)(node0 + r) * D + c);
        const float s = invc[r];
        sv.x *= s; sv.y *= s; sv.z *= s; sv.w *= s;
        *(float4*)&ldsA[r][c] = sv;
    }
    __syncthreads();

    const int wave = tid >> 5;
    const int lane = tid & 31;
    const int half = lane >> 4;   // K sub-pair selector per ISA A/B layout
    const int m    = lane & 15;   // row (A) / column (B) within tile
    const int nbase = wave * 64;

    v8f accR[4] = {};  // x @ Wr^T
    v8f accL[4] = {};  // aggr @ Wl^T

    for (int kt = 0; kt < 64; ++kt) {
        const int kk = kt * 4 + half * 2;
        const v2f ax = *(const v2f*)&ldsX[m][kk];
        const v2f ag = *(const v2f*)&ldsA[m][kk];
#pragma unroll
        for (int t = 0; t < 4; ++t) {
            const int n = nbase + t * 16 + m;
            const v2f bR = *(const v2f*)(Wr + (size_t)n * D + kk);
            const v2f bL = *(const v2f*)(Wl + (size_t)n * D + kk);
            accR[t] = __builtin_amdgcn_wmma_f32_16x16x4_f32(
                false, ax, false, bR, (short)0, accR[t], false, false);
            accL[t] = __builtin_amdgcn_wmma_f32_16x16x4_f32(
                false, ag, false, bL, (short)0, accL[t], false, false);
        }
    }

    // epilogue: C/D layout -> VGPR i holds M=i (lanes 0-15) / M=i+8 (lanes 16-31)
#pragma unroll
    for (int t = 0; t < 4; ++t) {
        const int col = nbase + t * 16 + m;
        const float bv = bias[col];
#pragma unroll
        for (int i = 0; i < 8; ++i) {
            const int row = node0 + i + half * 8;
            out[(size_t)row * D + col] = accR[t][i] + accL[t][i] + bv;
        }
    }
}

extern "C" void kernel_launch(void* const* d_in, const int* in_sizes, int n_in,
                              void* d_out, int out_size, void* d_ws, size_t ws_size,
                              hipStream_t stream) {
    const float*     x    = (const float*)d_in[0];
    const long long* ei   = (const long long*)d_in[1];   // int64 [2, E]
    const float*     Wl   = (const float*)d_in[2];
    const float*     Wr   = (const float*)d_in[3];
    const float*     br   = (const float*)d_in[4];
    float*           out  = (float*)d_out;

    const int d = 256;
    const int N = in_sizes[0] / d;   // 20000
    const int E = in_sizes[1] / 2;   // 320000

    float* sums   = (float*)d_ws;            // [N, 256]
    float* counts = sums + (size_t)N * d;    // [N]

    const int nzero = N * d + N;
    sage_zero_kernel<<<1024, 256, 0, stream>>>(sums, nzero);
    sage_count_kernel<<<1024, 256, 0, stream>>>(ei, counts, E);
    sage_scatter_kernel<<<4096, 256, 0, stream>>>(ei, x, sums, E);
    sage_gemm_kernel<<<N / 16, 128, 0, stream>>>(x, sums, counts, Wl, Wr, br, out);
}